// FaFMIMONet_256_32_32_31963146617589
// MI455X (gfx1250) — compile-verified
//
#include <hip/hip_runtime.h>

// Problem constants (from reference): A=5, B=2, K=5, C=13, H=W=256
#define A_ 5
#define B_ 2
#define K_ 5
#define C_ 13
#define H_ 256
#define W_ 256
#define LDSW 260          // padded LDS row stride in floats (1040 B = 65 * 16 B)
#define NT 1024           // 32 waves per workgroup
#define NCHUNK (H_ * (W_ / 4))   // 16384 float4 chunks per image

#if defined(__has_builtin)
#if __has_builtin(__builtin_amdgcn_global_load_async_to_lds_b128)
#define HAVE_ASYNC_LDS 1
#endif
#if __has_builtin(__builtin_amdgcn_s_wait_asynccnt)
#define HAVE_WAIT_ASYNC 1
#endif
#endif

#ifndef HAVE_ASYNC_LDS
#warning "CDNA5 probe: __builtin_amdgcn_global_load_async_to_lds_b128 NOT available; using sync staging fallback"
#endif
#ifndef HAVE_WAIT_ASYNC
#warning "CDNA5 probe: __builtin_amdgcn_s_wait_asynccnt NOT available; using inline asm s_wait_asynccnt"
#endif

// The async-to-LDS b128 builtin takes v4i pointers in AS1 (global) / AS3 (LDS)
typedef int v4i_t __attribute__((ext_vector_type(4)));
typedef __attribute__((address_space(1))) v4i_t gv4i_t;
typedef __attribute__((address_space(3))) v4i_t lv4i_t;

__global__ __launch_bounds__(NT)
void FaFMIMONet_fused_warp_kernel(const float* __restrict__ bevs,
                                  const float* __restrict__ trans,
                                  const int*   __restrict__ num_agent,
                                  const int*   __restrict__ center_agent_p,
                                  float*       __restrict__ out)
{
    extern __shared__ float smem[];   // H_ * LDSW floats = 266,240 bytes

    const int blk = blockIdx.x;       // 0 .. A*B*K*C-1 = 0..649
    const int c  = blk % C_;
    const int k  = (blk / C_) % K_;
    const int ab = blk / (C_ * K_);   // ab = a*B + b (bevs leading-dim order)
    const int a  = ab / B_;
    const int b  = ab % B_;
    const int tid = threadIdx.x;

    const size_t imgOff = ((size_t)(ab * K_ + k) * C_ + c) * (size_t)(H_ * W_);
    const float* gsrc = bevs + imgOff;
    float*       gdst = out  + imgOff;

    // ---- Stage source channel image (256KB) into LDS with padded rows ----
#if HAVE_ASYNC_LDS
    for (int ch = tid; ch < NCHUNK; ch += NT) {
        const int r  = ch >> 6;        // row (64 b128 chunks per row)
        const int cc = ch & 63;        // chunk within row
        __builtin_amdgcn_global_load_async_to_lds_b128(
            (gv4i_t*)(gsrc + (r * W_ + cc * 4)),
            (lv4i_t*)(smem + (r * LDSW + cc * 4)),
            0, 0);
    }
#if HAVE_WAIT_ASYNC
    __builtin_amdgcn_s_wait_asynccnt(0);
#else
    asm volatile("s_wait_asynccnt 0x0" ::: "memory");
#endif
    __syncthreads();
#else
    for (int ch = tid; ch < NCHUNK; ch += NT) {
        const int r  = ch >> 6;
        const int cc = ch & 63;
        const float4 v = *(const float4*)(gsrc + (r * W_ + cc * 4));
        *(float4*)(smem + (r * LDSW + cc * 4)) = v;
    }
    __syncthreads();
#endif

    // ---- Mask: out = warped if (a < n_agent[b]) && (a != center || k == K-1) else orig ----
    const int center = *center_agent_p;
    const int n      = num_agent[b * A_ + center];
    const bool doWarp = (a < n) && ((a != center) || (k == K_ - 1));

    if (!doWarp) {
        // pass-through copy (bit-exact): LDS -> global, coalesced b128
        for (int ch = tid; ch < NCHUNK; ch += NT) {
            const int r  = ch >> 6;
            const int cc = ch & 63;
            const float4 v = *(const float4*)(smem + (r * LDSW + cc * 4));
            *(float4*)(gdst + (r * W_ + cc * 4)) = v;
        }
        return;
    }

    // ---- Uniform affine params for this (a,k,b) ----
    const int tbase = (((a * K_) + k) * B_ + b) * 16;   // trans_mats[A][K][B][4][4]
    const float t00 = trans[tbase + 0];
    const float t01 = trans[tbase + 1];
    const float t10 = trans[tbase + 4];
    const float t11 = trans[tbase + 5];
    const float xt  =  4.0f * trans[tbase + 3] * (1.0f / 128.0f);
    const float yt  = -4.0f * trans[tbase + 7] * (1.0f / 128.0f);

    const float S = 2.0f / (float)W_;   // 0.0078125, exact

    // ---- Fused double bilinear warp: 16 LDS taps per output pixel ----
    for (int p = tid; p < H_ * W_; p += NT) {
        const int w = p & (W_ - 1);
        const int h = p >> 8;

        const float x = ((float)w + 0.5f) * S - 1.0f;
        const float y = ((float)h + 0.5f) * S - 1.0f;

        // translation warp source coordinate (in warped1 pixel space)
        const float ix = ((x + xt + 1.0f) * (float)W_ - 1.0f) * 0.5f;
        const float iy = ((y + yt + 1.0f) * (float)H_ - 1.0f) * 0.5f;
        const float fx0 = floorf(ix), fy0 = floorf(iy);
        const float wx = ix - fx0, wy = iy - fy0;
        const int x0 = (int)fx0, y0 = (int)fy0;

        float acc = 0.0f;
#pragma unroll
        for (int dy = 0; dy < 2; ++dy) {
#pragma unroll
            for (int dx = 0; dx < 2; ++dx) {
                const int qx = x0 + dx;
                const int qy = y0 + dy;
                const float wq = (dx ? wx : 1.0f - wx) * (dy ? wy : 1.0f - wy);
                const bool vq = (qx >= 0) & (qx < W_) & (qy >= 0) & (qy < H_);
                const int qxc = qx < 0 ? 0 : (qx > W_ - 1 ? W_ - 1 : qx);
                const int qyc = qy < 0 ? 0 : (qy > H_ - 1 ? H_ - 1 : qy);

                // warped1[qyc][qxc]: rotation-warp bilinear sample of the original
                const float xs = ((float)qxc + 0.5f) * S - 1.0f;
                const float ys = ((float)qyc + 0.5f) * S - 1.0f;
                const float rgx = t00 * xs + t01 * ys;
                const float rgy = t10 * xs + t11 * ys;
                const float rix = ((rgx + 1.0f) * (float)W_ - 1.0f) * 0.5f;
                const float riy = ((rgy + 1.0f) * (float)H_ - 1.0f) * 0.5f;
                const float rfx = floorf(rix), rfy = floorf(riy);
                const float rwx = rix - rfx, rwy = riy - rfy;
                const int rx0 = (int)rfx, ry0 = (int)rfy;

                float val = 0.0f;
#pragma unroll
                for (int ey = 0; ey < 2; ++ey) {
#pragma unroll
                    for (int ex = 0; ex < 2; ++ex) {
                        const int sx = rx0 + ex;
                        const int sy = ry0 + ey;
                        const float ws = (ex ? rwx : 1.0f - rwx) * (ey ? rwy : 1.0f - rwy);
                        const bool sv = (sx >= 0) & (sx < W_) & (sy >= 0) & (sy < H_);
                        const int sxc = sx < 0 ? 0 : (sx > W_ - 1 ? W_ - 1 : sx);
                        const int syc = sy < 0 ? 0 : (sy > H_ - 1 ? H_ - 1 : sy);
                        const float t = smem[syc * LDSW + sxc];
                        val += t * (sv ? ws : 0.0f);
                    }
                }
                acc += val * (vq ? wq : 0.0f);
            }
        }
        gdst[p] = acc;
    }
}

extern "C" void kernel_launch(void* const* d_in, const int* in_sizes, int n_in,
                              void* d_out, int out_size, void* d_ws, size_t ws_size,
                              hipStream_t stream) {
    (void)in_sizes; (void)n_in; (void)out_size; (void)d_ws; (void)ws_size;
    const float* bevs        = (const float*)d_in[0];   // (A*B, K, C, H, W) fp32
    const float* trans       = (const float*)d_in[1];   // (A, K, B, 4, 4) fp32
    const int*   num_agent   = (const int*)d_in[2];     // (B, A) int32
    // d_in[3] = batch_size (compile-time constant B_=2)
    const int*   center      = (const int*)d_in[4];     // scalar int32
    float*       out         = (float*)d_out;

    const dim3 grid(A_ * B_ * K_ * C_);   // 650 blocks, one per channel image
    const dim3 block(NT);
    const size_t ldsBytes = (size_t)H_ * LDSW * sizeof(float);  // 266,240 B < 320 KB

    hipLaunchKernelGGL(FaFMIMONet_fused_warp_kernel, grid, block, ldsBytes, stream,
                       bevs, trans, num_agent, center, out);
}